// CurveGrouping_57664230916170
// MI455X (gfx1250) — compile-verified
//
#include <hip/hip_runtime.h>

typedef float v2f __attribute__((ext_vector_type(2)));
typedef float v8f __attribute__((ext_vector_type(8)));

#define BATCH 4
#define NPTS  8192
#define KNN   16
#define CLEN  4
#define ROWS  128   // rows handled per workgroup
#define CCH   64    // column chunk size
#define DPAD  (CCH + 4)   // dist row stride in floats: 272 B, 16 B aligned

// ---------------------------------------------------------------------------
// Kernel 1: fused pairwise-distance (WMMA f32 16x16x4) + streaming top-16
// ---------------------------------------------------------------------------
__global__ __launch_bounds__(ROWS) void knn_kernel(const float* __restrict__ xyz,
                                                   int* __restrict__ knn)
{
    __shared__ float s_rows[ROWS][4];       // row xyz padded to 4
    __shared__ float s_rowsq[ROWS];         // |row|^2
    __shared__ float s_cols[CCH][4];        // column-chunk xyz padded to 4
    __shared__ float s_colsq[CCH];          // |col|^2
    __shared__ float s_dist[ROWS][DPAD];    // 16B-aligned rows for b128 scan

    const int tid     = threadIdx.x;
    const int b       = blockIdx.y;
    const int rowBase = blockIdx.x * ROWS;
    const float* xb   = xyz + (size_t)b * NPTS * 3;

    // Load this workgroup's 128 rows once (L2-resident source).
    {
        const int r = rowBase + tid;
        const float x = xb[r * 3 + 0];
        const float y = xb[r * 3 + 1];
        const float z = xb[r * 3 + 2];
        s_rows[tid][0] = x; s_rows[tid][1] = y;
        s_rows[tid][2] = z; s_rows[tid][3] = 0.0f;
        s_rowsq[tid]   = x * x + y * y + z * z;
    }
    __syncthreads();

    const int lane = tid & 31;
    const int wave = tid >> 5;
    const int half = lane >> 4;   // 0: lanes 0-15, 1: lanes 16-31
    const int lm   = lane & 15;
    const int k0   = half << 1;   // K base held by this lane-half (A & B frags)

    // Chunk-invariant preloads: A fragments + the 16 row-norms this thread
    // needs in the d2 epilogue. Stays in registers for all 128 chunks.
    v2f afrag[2];
    float rsq[2][8];
#pragma unroll
    for (int t = 0; t < 2; ++t) {
        const int tr = wave * 2 + t;
        afrag[t].x = s_rows[tr * 16 + lm][k0];
        afrag[t].y = s_rows[tr * 16 + lm][k0 + 1];
#pragma unroll
        for (int r = 0; r < 8; ++r)
            rsq[t][r] = s_rowsq[tr * 16 + r + 8 * half];
    }

    // Per-thread sorted top-16 (ascending; [KNN-1] is current worst).
    float bestd[KNN];
    int   besti[KNN];
#pragma unroll
    for (int s = 0; s < KNN; ++s) { bestd[s] = 3.4e38f; besti[s] = 0; }
    float dmax = bestd[KNN - 1];

    for (int cb = 0; cb < NPTS; cb += CCH) {
        __syncthreads();          // previous chunk's scan done before reload
        if (tid < CCH) {
            const int c = cb + tid;
            const float x = xb[c * 3 + 0];
            const float y = xb[c * 3 + 1];
            const float z = xb[c * 3 + 2];
            s_cols[tid][0] = x; s_cols[tid][1] = y;
            s_cols[tid][2] = z; s_cols[tid][3] = 0.0f;
            s_colsq[tid]   = x * x + y * y + z * z;
        }
        __syncthreads();

        // 32 tiles of 16x16; each of the 4 waves computes 2 row-tiles x 4 col-tiles.
#pragma unroll
        for (int t = 0; t < 2; ++t) {
            const int tr = wave * 2 + t;
#pragma unroll
            for (int tc = 0; tc < 4; ++tc) {
                // B fragment (4x16 f32): lane holds N=lm, K={k0,k0+1}
                v2f bm;
                bm.x = s_cols[tc * 16 + lm][k0];
                bm.y = s_cols[tc * 16 + lm][k0 + 1];
                v8f c8 = {};
                // D = A x B + 0  -> dot(row_M, col_N)
                c8 = __builtin_amdgcn_wmma_f32_16x16x4_f32(
                        /*neg_a=*/false, afrag[t], /*neg_b=*/false, bm,
                        /*c_mod=*/(short)0, c8,
                        /*reuse_a=*/false, /*reuse_b=*/false);
                const float sc = s_colsq[tc * 16 + lm];
                // C/D layout: lane half selects M offset (+8), VGPR r selects row
#pragma unroll
                for (int r = 0; r < 8; ++r) {
                    const int m = tr * 16 + r + 8 * half;
                    s_dist[m][tc * 16 + lm] = rsq[t][r] + sc - 2.0f * c8[r];
                }
            }
        }
        __syncthreads();

        // Streaming selection: thread tid owns row tid. float4 loads (b128),
        // single min-guard for 4 candidates; inserts stay ascending-j stable.
        const float4* drow = reinterpret_cast<const float4*>(&s_dist[tid][0]);
        for (int j4 = 0; j4 < CCH / 4; ++j4) {
            const float4 v = drow[j4];
            const float m4 = fminf(fminf(v.x, v.y), fminf(v.z, v.w));
            if (m4 < dmax) {
                const float vv[4] = { v.x, v.y, v.z, v.w };
#pragma unroll
                for (int q = 0; q < 4; ++q) {
                    const float dv = vv[q];
                    if (dv < dmax) {
                        float cd = dv;
                        int   ci = cb + j4 * 4 + q;
#pragma unroll
                        for (int s = 0; s < KNN; ++s) {
                            const bool  p  = cd < bestd[s];  // strict: tie-stable
                            const float td = bestd[s];
                            const int   ti = besti[s];
                            if (p) { bestd[s] = cd; besti[s] = ci; cd = td; ci = ti; }
                        }
                        dmax = bestd[KNN - 1];
                    }
                }
            }
        }
    }

    // Emit kNN table (rank order == jax.lax.top_k order: ascending distance).
    const int row = rowBase + tid;
    int* dst = knn + ((size_t)b * NPTS + row) * KNN;
#pragma unroll
    for (int s = 0; s < KNN; ++s) dst[s] = besti[s];
}

// ---------------------------------------------------------------------------
// Kernel 2: 3-step random walk through the kNN graph
// ---------------------------------------------------------------------------
__global__ void walk_kernel(const int* __restrict__ knn,
                            const int* __restrict__ rk,
                            float* __restrict__ out)
{
    const int gid = blockIdx.x * blockDim.x + threadIdx.x;
    if (gid >= BATCH * NPTS) return;
    const int b = gid / NPTS;
    const int n = gid % NPTS;

    int cur = n;
    float* o = out + (size_t)gid * CLEN;
    o[0] = (float)cur;
#pragma unroll
    for (int s = 0; s < CLEN - 1; ++s) {
        const int r = rk[((size_t)s * BATCH + b) * NPTS + n];  // in [1, K)
        cur = knn[((size_t)b * NPTS + cur) * KNN + r];
        o[s + 1] = (float)cur;
    }
}

// ---------------------------------------------------------------------------
extern "C" void kernel_launch(void* const* d_in, const int* in_sizes, int n_in,
                              void* d_out, int out_size, void* d_ws, size_t ws_size,
                              hipStream_t stream)
{
    const float* xyz    = (const float*)d_in[0];  // (B, N, 3) f32
    const int*   rand_k = (const int*)d_in[1];    // (CLEN-1, B, N) i32
    int*   knn = (int*)d_ws;                      // (B, N, K) i32 scratch, 2 MB
    float* out = (float*)d_out;                   // (B, N, CLEN)

    dim3 grid1(NPTS / ROWS, BATCH);               // 64 x 4 workgroups
    knn_kernel<<<grid1, ROWS, 0, stream>>>(xyz, knn);

    const int total = BATCH * NPTS;
    walk_kernel<<<(total + 255) / 256, 256, 0, stream>>>(knn, rand_k, out);
}